// GraphConvNet_19533511262573
// MI455X (gfx1250) — compile-verified
//
#include <hip/hip_runtime.h>

// ---------------------------------------------------------------------------
// GCN forward on MI455X (gfx1250, wave32).
//
//   h1   = x @ W1                       (WMMA f32 16x16x4, LDS-staged operands)
//   agg1 = Dinv (A) Dinv @ h1           (edge scatter-add, L2-resident atomics)
//   p    = maxpool2(lrelu(agg1 + self + b1))
//   h2   = p @ W2                       (fused into epilogue 1)
//   agg2 = Dinv (A) Dinv @ h2
//   out  = lrelu(agg2 + self + b2) @ Wl + bl
//
// All node-indexed arrays (<= 8 MB each) stay resident in the 192 MB L2, so
// random gathers and scatter atomics are L2 RMWs; HBM only streams x and the
// int64 edge list.
// ---------------------------------------------------------------------------

typedef __attribute__((ext_vector_type(2))) float v2f;
typedef __attribute__((ext_vector_type(8))) float v8f;

#if defined(__gfx1250__) && __has_builtin(__builtin_amdgcn_global_load_async_to_lds_b32)
#define USE_ASYNC_LDS 1
typedef __attribute__((address_space(1))) int as1_int;  // global (AS1) int*
typedef __attribute__((address_space(3))) int as3_int;  // LDS (AS3) int*
#endif

__device__ __forceinline__ float lrelu(float v) {
  return v > 0.0f ? v : 0.01f * v;
}

// --- 1) zero accumulators, deg = 1 (self-loop) ------------------------------
__global__ void k_init(float* __restrict__ agg1, float* __restrict__ agg2,
                       float* __restrict__ deg, int N) {
  int i = blockIdx.x * blockDim.x + threadIdx.x;
  if (i < 20 * N) agg1[i] = 0.0f;
  if (i < 5 * N)  agg2[i] = 0.0f;
  if (i < N)      deg[i]  = 1.0f;
}

// --- 2) deg[dst] += 1 per edge ----------------------------------------------
__global__ void k_degree(const long long* __restrict__ ei, int E,
                         float* __restrict__ deg) {
  int e = blockIdx.x * blockDim.x + threadIdx.x;
  if (e < E) atomicAdd(&deg[(int)ei[(size_t)E + e]], 1.0f);
}

// --- 3) deg -> rsqrt(deg) in place ------------------------------------------
__global__ void k_rsqrt(float* __restrict__ deg, int N) {
  int i = blockIdx.x * blockDim.x + threadIdx.x;
  if (i < N) deg[i] = rsqrtf(deg[i]);
}

// --- 4) h1 = x @ W1 via V_WMMA_F32_16X16X4_F32, LDS-staged ------------------
// Block = 256 threads = 8 waves. Block tile: 64 rows x 20(->32) cols, K=30->32.
// LDS holds x tile [64][32] and W1 [32][32], both zero-padded, so every WMMA
// fragment load is an unconditional ds_load (no exec-mask churn, EXEC==all-1s
// at each wmma as the ISA requires).
// Fragment layouts (ISA 7.12.2): A lane=M (mod 16), VGPR pair = K, lanes 16-31
// carry K+2; B mirrored with lane=N; C/D VGPR j holds M=j+8*(lane>=16).
__global__ void k_gemm1_wmma(const float* __restrict__ x,
                             const float* __restrict__ W1,
                             float* __restrict__ h1, int N) {
  __shared__ float wsm[32 * 32];  // W1 padded [K=32][N=32]
  __shared__ float xs[64 * 32];   // x tile   [row=64][K=32]

  const int t = threadIdx.x;
  const int row0 = blockIdx.x * 64;

  // Zero-fill both tiles (padding cells must be 0).
  for (int idx = t; idx < 1024; idx += 256) wsm[idx] = 0.0f;
  for (int idx = t; idx < 2048; idx += 256) xs[idx] = 0.0f;
  __syncthreads();

  // Stage W1 (30x20 -> padded [k][n]).
  for (int idx = t; idx < 600; idx += 256)
    wsm[(idx / 20) * 32 + (idx % 20)] = W1[idx];

  // Stage x tile: flat range [row0*30, +1920), scattered to 32-float rows.
  const size_t base = (size_t)row0 * 30;
  const int nval = min(1920, N * 30 - (int)base);
  for (int idx = t; idx < nval; idx += 256) {
    const int dst = (idx / 30) * 32 + (idx % 30);
#if USE_ASYNC_LDS
    // gfx1250 async global->LDS copy (ASYNCcnt path).
    __builtin_amdgcn_global_load_async_to_lds_b32(
        (as1_int*)(unsigned long long)(const char*)(x + base + idx),
        (as3_int*)(unsigned int)(unsigned long long)(char*)&xs[dst],
        0, 0);
#else
    xs[dst] = x[base + idx];
#endif
  }
#if USE_ASYNC_LDS
#if __has_builtin(__builtin_amdgcn_s_wait_asynccnt)
  __builtin_amdgcn_s_wait_asynccnt(0);
#else
  asm volatile("s_wait_asynccnt 0" ::: "memory");
#endif
#endif
  __syncthreads();

  const int lane = t & 31, wave = t >> 5;
  const int rowTile = wave >> 1, colTile = wave & 1;
  const int lm = lane & 15, lh = lane >> 4;

  const float* __restrict__ arow = &xs[(rowTile * 16 + lm) * 32];  // A row
  const float* __restrict__ bcol = &wsm[colTile * 16 + lm];        // B column

  v8f c = {};
#pragma unroll
  for (int ks = 0; ks < 8; ++ks) {
    const int k0 = ks * 4 + lh * 2;
    v2f a = *(const v2f*)(arow + k0);  // 8B-aligned ds_load_b64, no guard
    v2f b;
    b.x = bcol[(k0 + 0) * 32];
    b.y = bcol[(k0 + 1) * 32];
    c = __builtin_amdgcn_wmma_f32_16x16x4_f32(false, a, false, b,
                                              (short)0, c, false, false);
  }

  const int n = colTile * 16 + lm;
  const int rbase = row0 + rowTile * 16 + 8 * lh;
  if (n < 20) {
#pragma unroll
    for (int j = 0; j < 8; ++j) {
      const int rr = rbase + j;
      if (rr < N) h1[(size_t)rr * 20 + n] = c[j];
    }
  }
}

// --- 5) agg1[dst] += h1[src] * dinv[src]*dinv[dst] --------------------------
// Grid-stride with software prefetch of next iteration's gathered row: the
// index->gather dependent chain is the one latency occupancy can't hide.
__global__ void k_scatter1(const long long* __restrict__ ei, int E,
                           const float* __restrict__ dinv,
                           const float* __restrict__ h1,
                           float* __restrict__ agg1) {
  const int stride = gridDim.x * blockDim.x;
  for (int e = blockIdx.x * blockDim.x + threadIdx.x; e < E; e += stride) {
    const int en = e + stride;
    if (en < E) {
      const int sn = (int)ei[en];
      __builtin_prefetch(h1 + (size_t)sn * 20, 0, 3);  // global_prefetch_b8
    }
    const int s = (int)ei[e];
    const int d = (int)ei[(size_t)E + e];
    const float norm = dinv[s] * dinv[d];
    const float4* __restrict__ hp = (const float4*)(h1 + (size_t)s * 20);
    float* __restrict__ op = agg1 + (size_t)d * 20;
#pragma unroll
    for (int q = 0; q < 5; ++q) {
      float4 v = hp[q];
      atomicAdd(op + 4 * q + 0, v.x * norm);
      atomicAdd(op + 4 * q + 1, v.y * norm);
      atomicAdd(op + 4 * q + 2, v.z * norm);
      atomicAdd(op + 4 * q + 3, v.w * norm);
    }
  }
}

// --- 6) fused: + self-loop + b1, lrelu, maxpool2, @W2 -> h2 -----------------
__global__ void k_epilogue1(const float* __restrict__ agg1,
                            const float* __restrict__ h1,
                            const float* __restrict__ dinv,
                            const float* __restrict__ b1,
                            const float* __restrict__ W2,
                            float* __restrict__ h2, int N) {
  int i = blockIdx.x * blockDim.x + threadIdx.x;
  if (i >= N) return;
  const float di = dinv[i];
  const float self = di * di;
  const float4* __restrict__ ap = (const float4*)(agg1 + (size_t)i * 20);
  const float4* __restrict__ hp = (const float4*)(h1 + (size_t)i * 20);
  float p[10];
#pragma unroll
  for (int q = 0; q < 5; ++q) {
    float4 a = ap[q], h = hp[q];
    float v0 = lrelu(a.x + h.x * self + b1[4 * q + 0]);
    float v1 = lrelu(a.y + h.y * self + b1[4 * q + 1]);
    float v2 = lrelu(a.z + h.z * self + b1[4 * q + 2]);
    float v3 = lrelu(a.w + h.w * self + b1[4 * q + 3]);
    p[2 * q]     = fmaxf(v0, v1);
    p[2 * q + 1] = fmaxf(v2, v3);
  }
#pragma unroll
  for (int k = 0; k < 5; ++k) {
    float acc = 0.0f;
#pragma unroll
    for (int j = 0; j < 10; ++j) acc += p[j] * W2[j * 5 + k];
    h2[(size_t)i * 5 + k] = acc;
  }
}

// --- 7) agg2[dst] += h2[src] * dinv[src]*dinv[dst] --------------------------
__global__ void k_scatter2(const long long* __restrict__ ei, int E,
                           const float* __restrict__ dinv,
                           const float* __restrict__ h2,
                           float* __restrict__ agg2) {
  const int stride = gridDim.x * blockDim.x;
  for (int e = blockIdx.x * blockDim.x + threadIdx.x; e < E; e += stride) {
    const int en = e + stride;
    if (en < E) {
      const int sn = (int)ei[en];
      __builtin_prefetch(h2 + (size_t)sn * 5, 0, 3);
    }
    const int s = (int)ei[e];
    const int d = (int)ei[(size_t)E + e];
    const float norm = dinv[s] * dinv[d];
    const float* __restrict__ hp = h2 + (size_t)s * 5;
    float* __restrict__ op = agg2 + (size_t)d * 5;
#pragma unroll
    for (int j = 0; j < 5; ++j) atomicAdd(op + j, hp[j] * norm);
  }
}

// --- 8) fused: + self-loop + b2, lrelu, @Wl + bl -> out ---------------------
__global__ void k_epilogue2(const float* __restrict__ agg2,
                            const float* __restrict__ h2,
                            const float* __restrict__ dinv,
                            const float* __restrict__ b2,
                            const float* __restrict__ Wl,
                            const float* __restrict__ bl,
                            float* __restrict__ out, int N) {
  int i = blockIdx.x * blockDim.x + threadIdx.x;
  if (i >= N) return;
  const float di = dinv[i];
  const float self = di * di;
  float v[5];
#pragma unroll
  for (int k = 0; k < 5; ++k)
    v[k] = lrelu(agg2[(size_t)i * 5 + k] + h2[(size_t)i * 5 + k] * self + b2[k]);
  float o0 = bl[0], o1 = bl[1];
#pragma unroll
  for (int k = 0; k < 5; ++k) {
    o0 += v[k] * Wl[k * 2 + 0];
    o1 += v[k] * Wl[k * 2 + 1];
  }
  out[(size_t)i * 2 + 0] = o0;
  out[(size_t)i * 2 + 1] = o1;
}

// ---------------------------------------------------------------------------
extern "C" void kernel_launch(void* const* d_in, const int* in_sizes, int n_in,
                              void* d_out, int out_size, void* d_ws, size_t ws_size,
                              hipStream_t stream) {
  const float*     x  = (const float*)d_in[0];
  const long long* ei = (const long long*)d_in[1];  // int64 edge_index [2, E]
  const float*     W1 = (const float*)d_in[2];
  const float*     b1 = (const float*)d_in[3];
  const float*     W2 = (const float*)d_in[4];
  const float*     b2 = (const float*)d_in[5];
  const float*     Wl = (const float*)d_in[6];
  const float*     bl = (const float*)d_in[7];
  float* out = (float*)d_out;

  const int N = in_sizes[0] / 30;
  const int E = in_sizes[1] / 2;

  // Workspace carve-out (256B-aligned): dinv N | h1 20N | agg1 20N | h2 5N | agg2 5N
  char* ws = (char*)d_ws;
  size_t o = 0;
  auto bump = [&](size_t bytes) {
    size_t cur = o;
    o = (o + bytes + 255) & ~(size_t)255;
    return cur;
  };
  float* deg  = (float*)(ws + bump((size_t)N * 4));        // becomes dinv
  float* h1   = (float*)(ws + bump((size_t)N * 20 * 4));
  float* agg1 = (float*)(ws + bump((size_t)N * 20 * 4));
  float* h2   = (float*)(ws + bump((size_t)N * 5 * 4));
  float* agg2 = (float*)(ws + bump((size_t)N * 5 * 4));
  (void)ws_size; (void)n_in; (void)out_size;

  const int T = 256;
  const int edgeBlocks = (E / 4 + T - 1) / T;  // 4 edges per thread (grid-stride)
  k_init      <<<(20 * N + T - 1) / T, T, 0, stream>>>(agg1, agg2, deg, N);
  k_degree    <<<(E + T - 1) / T,      T, 0, stream>>>(ei, E, deg);
  k_rsqrt     <<<(N + T - 1) / T,      T, 0, stream>>>(deg, N);
  k_gemm1_wmma<<<(N + 63) / 64,        T, 0, stream>>>(x, W1, h1, N);
  k_scatter1  <<<edgeBlocks,           T, 0, stream>>>(ei, E, deg, h1, agg1);
  k_epilogue1 <<<(N + T - 1) / T,      T, 0, stream>>>(agg1, h1, deg, b1, W2, h2, N);
  k_scatter2  <<<edgeBlocks,           T, 0, stream>>>(ei, E, deg, h2, agg2);
  k_epilogue2 <<<(N + T - 1) / T,      T, 0, stream>>>(agg2, h2, deg, b2, Wl, bl, out, N);
}